// TransitionDown_16544214024413
// MI455X (gfx1250) — compile-verified
//
#include <hip/hip_runtime.h>
#include <math.h>

#define N_PTS   32768
#define C_IN    128
#define C_OUT   256
#define KNN_K   16
#define M_CL    8192
#define BN_EPS  1e-5f

// Points cached in LDS for FPS: 26*1024 points * 3 floats = 319,488 bytes
#define FPS_LDS_PTS (26 * 1024)

typedef __attribute__((ext_vector_type(2))) float v2f;
typedef __attribute__((ext_vector_type(8))) float v8f;

// ---------------------------------------------------------------------------
// 1) Farthest point sampling: single persistent workgroup, 1024 threads.
//    - 26624 of 32768 points cached in LDS (SoA, bank-conflict-free for the
//      p = t + j*1024 access pattern); remaining 6144 stream from L2.
//    - min_d lives in registers (32 floats/thread).
//    - ONE barrier per iteration: xor-butterfly leaves the wave-reduced
//      argmax in every lane; all waves redundantly reduce the 32 LDS
//      partials, so no broadcast barrier is needed. Double-buffered
//      partial arrays remove the WAR hazard.
// ---------------------------------------------------------------------------
__global__ __launch_bounds__(1024) void fps_kernel(const float* __restrict__ pos,
                                                   int* __restrict__ id_out)
{
    const int t    = threadIdx.x;      // 0..1023
    const int lane = t & 31;

    __shared__ float s_px[FPS_LDS_PTS];
    __shared__ float s_py[FPS_LDS_PTS];
    __shared__ float s_pz[FPS_LDS_PTS];
    __shared__ float s_val[2][32];
    __shared__ int   s_idx[2][32];

    // cooperative SoA fill of the LDS point cache
    for (int i = t; i < FPS_LDS_PTS; i += 1024) {
        s_px[i] = pos[3 * i + 0];
        s_py[i] = pos[3 * i + 1];
        s_pz[i] = pos[3 * i + 2];
    }

    float mind[32];
#pragma unroll
    for (int j = 0; j < 32; ++j) mind[j] = 3.4e38f;

    if (t == 0) id_out[0] = 0;
    __syncthreads();
    int last = 0;

    for (int it = 1; it < M_CL; ++it) {
        float lx, ly, lz;
        if (last < FPS_LDS_PTS) {
            lx = s_px[last]; ly = s_py[last]; lz = s_pz[last];
        } else {
            lx = pos[3 * last + 0]; ly = pos[3 * last + 1]; lz = pos[3 * last + 2];
        }

        float best = -1.0f;
        int   bidx = 0;
#pragma unroll
        for (int j = 0; j < 32; ++j) {
            const int p = t + (j << 10);            // t + j*1024, covers 32768
            float px, py, pz;
            if (j < 26) {                           // compile-time resolved branch
                px = s_px[p]; py = s_py[p]; pz = s_pz[p];
            } else {
                px = pos[3 * p + 0]; py = pos[3 * p + 1]; pz = pos[3 * p + 2];
            }
            const float dx = px - lx;
            const float dy = py - ly;
            const float dz = pz - lz;
            const float d  = dx * dx + dy * dy + dz * dz;
            const float m  = fminf(mind[j], d);
            mind[j] = m;
            if (m > best) { best = m; bidx = p; }   // strict > keeps lowest p in-thread
        }
        // wave32 argmax butterfly, lower index wins ties (matches jnp.argmax)
#pragma unroll
        for (int off = 16; off > 0; off >>= 1) {
            const float ov = __shfl_xor(best, off, 32);
            const int   oi = __shfl_xor(bidx, off, 32);
            if (ov > best || (ov == best && oi < bidx)) { best = ov; bidx = oi; }
        }
        const int buf = it & 1;
        if (lane == 0) { s_val[buf][t >> 5] = best; s_idx[buf][t >> 5] = bidx; }
        __syncthreads();
        // every wave reduces the 32 partials; butterfly leaves result in all lanes
        float v = s_val[buf][lane];
        int   i = s_idx[buf][lane];
#pragma unroll
        for (int off = 16; off > 0; off >>= 1) {
            const float ov = __shfl_xor(v, off, 32);
            const int   oi = __shfl_xor(i, off, 32);
            if (ov > v || (ov == v && oi < i)) { v = ov; i = oi; }
        }
        last = i;                                   // identical in every thread
        if (t == 0) id_out[it] = i;
    }
}

// ---------------------------------------------------------------------------
// 2) Gather sub_pos / sub_batch (batch is all zero in the reference).
// ---------------------------------------------------------------------------
__global__ __launch_bounds__(256) void gather_sub(const float* __restrict__ pos,
                                                  const int* __restrict__ id,
                                                  float* __restrict__ sub_pos,
                                                  int* __restrict__ sub_batch)
{
    const int m = blockIdx.x * blockDim.x + threadIdx.x;
    if (m < M_CL) {
        const int p = id[m];
        sub_pos[3 * m + 0] = pos[3 * p + 0];
        sub_pos[3 * m + 1] = pos[3 * p + 1];
        sub_pos[3 * m + 2] = pos[3 * p + 2];
        sub_batch[m] = 0;
    }
}

// ---------------------------------------------------------------------------
// 3) KNN: one thread per query, 16-entry register top-k with unrolled
//    replace-max insertion (static register indexing only).
// ---------------------------------------------------------------------------
__global__ __launch_bounds__(256) void knn_kernel(const float* __restrict__ pos,
                                                  const int* __restrict__ id,
                                                  int* __restrict__ nbr)
{
    const int m = blockIdx.x * blockDim.x + threadIdx.x;   // 0..8191
    const int q = id[m];
    const float qx = pos[3 * q + 0];
    const float qy = pos[3 * q + 1];
    const float qz = pos[3 * q + 2];
    const float qq = qx * qx + qy * qy + qz * qz;

    float bd[KNN_K];
    int   bi[KNN_K];
#pragma unroll
    for (int i = 0; i < KNN_K; ++i) { bd[i] = 3.4e38f; bi[i] = 0; }
    float wmax = 3.4e38f;

    for (int n = 0; n < N_PTS; ++n) {
        const float px = pos[3 * n + 0];
        const float py = pos[3 * n + 1];
        const float pz = pos[3 * n + 2];
        // match reference formula: |q|^2 - 2 q.p + |p|^2
        const float d = qq - 2.0f * (qx * px + qy * py + qz * pz)
                        + (px * px + py * py + pz * pz);
        if (d < wmax) {
            bool done = false;
#pragma unroll
            for (int i = 0; i < KNN_K; ++i) {
                if (!done && bd[i] == wmax) { bd[i] = d; bi[i] = n; done = true; }
            }
            wmax = -3.4e38f;
#pragma unroll
            for (int i = 0; i < KNN_K; ++i) wmax = fmaxf(wmax, bd[i]);
        }
    }
#pragma unroll
    for (int i = 0; i < KNN_K; ++i) nbr[m * KNN_K + i] = bi[i];
}

// ---------------------------------------------------------------------------
// 4) GEMM h = x @ W + b with V_WMMA_F32_16X16X4_F32.
//    Block = 256 threads = 8 waves; wave w owns row-tile (blockIdx.x*8+w)*16,
//    col-tile blockIdx.y*16; K-loop: 32 WMMAs over C_IN=128.
//    A 16x4 f32 layout (ISA 7.12.2): lanes 0-15 rows, VGPR{0,1} = K{0,1}
//    (lanes 0-15) / K{2,3} (lanes 16-31). B 4x16 mirrored.
// ---------------------------------------------------------------------------
__global__ __launch_bounds__(256) void gemm_bias_wmma(const float* __restrict__ x,
                                                      const float* __restrict__ W,
                                                      const float* __restrict__ bias,
                                                      float* __restrict__ h)
{
    const int lane = threadIdx.x & 31;
    const int wave = threadIdx.x >> 5;              // 0..7
    const int r0   = (blockIdx.x * 8 + wave) * 16;  // row tile origin
    const int c0   = blockIdx.y * 16;               // col tile origin
    const int lm   = lane & 15;                     // row/col within tile
    const int lh   = lane >> 4;                     // half-wave select (K 0/1 vs 2/3)

    v8f acc = {};
    const float* xrow = x + (size_t)(r0 + lm) * C_IN;
#pragma unroll
    for (int k = 0; k < C_IN; k += 4) {
        const int ka = k + lh * 2;
        v2f a;
        a.x = xrow[ka];
        a.y = xrow[ka + 1];
        v2f b;
        b.x = W[(size_t)ka * C_OUT + c0 + lm];
        b.y = W[(size_t)(ka + 1) * C_OUT + c0 + lm];
        acc = __builtin_amdgcn_wmma_f32_16x16x4_f32(false, a, false, b,
                                                    (short)0, acc, false, false);
    }

    const float bv = bias[c0 + lm];
#pragma unroll
    for (int v = 0; v < 8; ++v) {
        const int row = r0 + v + lh * 8;            // D layout: VGPR v -> M=v / M=v+8
        h[(size_t)row * C_OUT + c0 + lm] = acc[v] + bv;
    }
}

// ---------------------------------------------------------------------------
// 5) BatchNorm stats, deterministic two-pass (no float atomics).
// ---------------------------------------------------------------------------
__global__ __launch_bounds__(256) void bn_partial(const float* __restrict__ h,
                                                  float* __restrict__ psum,
                                                  float* __restrict__ psumsq)
{
    const int c  = threadIdx.x;            // channel
    const int rb = blockIdx.x * 128;       // 256 blocks x 128 rows
    float s = 0.0f, s2 = 0.0f;
    for (int j = 0; j < 128; ++j) {
        const float v = h[(size_t)(rb + j) * C_OUT + c];
        s  += v;
        s2 += v * v;
    }
    psum[blockIdx.x * C_OUT + c]   = s;
    psumsq[blockIdx.x * C_OUT + c] = s2;
}

__global__ __launch_bounds__(256) void bn_finalize(const float* __restrict__ psum,
                                                   const float* __restrict__ psumsq,
                                                   float* __restrict__ mu,
                                                   float* __restrict__ rsig)
{
    const int c = threadIdx.x;
    float s = 0.0f, s2 = 0.0f;
    for (int b = 0; b < 256; ++b) {        // fixed order -> deterministic
        s  += psum[b * C_OUT + c];
        s2 += psumsq[b * C_OUT + c];
    }
    const float m   = s / (float)N_PTS;
    const float var = s2 / (float)N_PTS - m * m;   // biased variance
    mu[c]   = m;
    rsig[c] = rsqrtf(var + BN_EPS);
}

// ---------------------------------------------------------------------------
// 6) Per-cluster max over K neighbors of BN+ReLU(h). Apply the full transform
//    per element before max (correct even for negative gamma).
// ---------------------------------------------------------------------------
__global__ __launch_bounds__(256) void cluster_max(const float* __restrict__ h,
                                                   const int* __restrict__ nbr,
                                                   const float* __restrict__ mu,
                                                   const float* __restrict__ rsig,
                                                   const float* __restrict__ gamma,
                                                   const float* __restrict__ beta,
                                                   float* __restrict__ out)
{
    const int c = threadIdx.x;
    const int m = blockIdx.x;
    const float mu_c = mu[c];
    const float rs   = rsig[c];
    const float g    = gamma[c];
    const float be   = beta[c];
    float mx = -3.4e38f;
#pragma unroll
    for (int k = 0; k < KNN_K; ++k) {
        const int idx = nbr[m * KNN_K + k];
        float v = h[(size_t)idx * C_OUT + c];
        v = g * (v - mu_c) * rs + be;
        v = fmaxf(v, 0.0f);
        mx = fmaxf(mx, v);
    }
    out[(size_t)m * C_OUT + c] = mx;
}

// ---------------------------------------------------------------------------
// Launch
// ---------------------------------------------------------------------------
extern "C" void kernel_launch(void* const* d_in, const int* in_sizes, int n_in,
                              void* d_out, int out_size, void* d_ws, size_t ws_size,
                              hipStream_t stream)
{
    const float* x     = (const float*)d_in[0];   // [N, C_IN]
    const float* pos   = (const float*)d_in[1];   // [N, 3]
    // d_in[2] = batch (all zeros, unused)
    const float* W     = (const float*)d_in[3];   // [C_IN, C_OUT]
    const float* b     = (const float*)d_in[4];   // [C_OUT]
    const float* gamma = (const float*)d_in[5];   // [C_OUT]
    const float* beta  = (const float*)d_in[6];   // [C_OUT]

    char* ws = (char*)d_ws;
    float* h    = (float*)(ws + 0);                         // 32768*256*4 = 33554432
    int*   id   = (int*)  (ws + 33554432);                  // 8192*4
    int*   nbr  = (int*)  (ws + 33587200);                  // 8192*16*4
    float* psum = (float*)(ws + 34111488);                  // 256*256*4
    float* psq  = (float*)(ws + 34373632);                  // 256*256*4
    float* mu   = (float*)(ws + 34635776);                  // 256*4
    float* rsig = (float*)(ws + 34636800);                  // 256*4

    float* out       = (float*)d_out;                       // [8192, 256]
    float* sub_pos   = out + (size_t)M_CL * C_OUT;          // [8192, 3]
    int*   sub_batch = (int*)(sub_pos + (size_t)M_CL * 3);  // [8192]

    fps_kernel   <<<1, 1024, 0, stream>>>(pos, id);
    gather_sub   <<<M_CL / 256, 256, 0, stream>>>(pos, id, sub_pos, sub_batch);
    knn_kernel   <<<M_CL / 256, 256, 0, stream>>>(pos, id, nbr);
    gemm_bias_wmma<<<dim3(N_PTS / 128, C_OUT / 16), 256, 0, stream>>>(x, W, b, h);
    bn_partial   <<<N_PTS / 128, 256, 0, stream>>>(h, psum, psq);
    bn_finalize  <<<1, 256, 0, stream>>>(psum, psq, mu, rsig);
    cluster_max  <<<M_CL, 256, 0, stream>>>(h, nbr, mu, rsig, gamma, beta, out);
}